// RoiPoolingConv_54425825575042
// MI455X (gfx1250) — compile-verified
//
#include <hip/hip_runtime.h>

// ROI bilinear pooling (crop-and-resize) on (1,192,192,48,64) fp32 volume,
// 64 ROIs -> (1,64,7,7,64). Expressed as two fp32 WMMA GEMMs per ROI
// (separable bilinear resize = WX * crop * WY^T), using V_WMMA_F32_16X16X4_F32
// to keep full fp32 precision on the CDNA5 matrix pipe.

typedef __attribute__((ext_vector_type(2))) float v2f;
typedef __attribute__((ext_vector_type(8))) float v8f;

#define XD 192
#define YD 192
#define ZD 48
#define CD 64
#define PP 7

__device__ __forceinline__ int clampi(int v, int lo, int hi) {
  return v < lo ? lo : (v > hi ? hi : v);
}

// Corner coordinate for axis sample index k (k even -> low corner, odd -> high),
// matching the reference: s = p*(len/7), i0=floor(s), clamp into crop then volume.
__device__ __forceinline__ int axcoord(int k, int len, int base, int maxd) {
  int kk = k > 13 ? 13 : k;
  int p = kk >> 1;
  float s = (float)p * ((float)len / 7.0f);
  float fl = floorf(s);
  int i0 = (int)fl;
  int a = (kk & 1) ? clampi(i0 + 1, 0, len - 1) : clampi(i0, 0, len - 1);
  return clampi(base + a, 0, maxd - 1);
}

// Interp-matrix element W[m][k]: row m has exactly two nonzeros at k=2m (1-f)
// and k=2m+1 (f). Rows m>=7 and cols k>=14 are zero padding.
__device__ __forceinline__ float axweight(int m, int k, int len) {
  if (m >= PP || k >= 14 || (k >> 1) != m) return 0.0f;
  float s = (float)m * ((float)len / 7.0f);
  float f = s - floorf(s);
  return (k & 1) ? f : (1.0f - f);
}

__global__ __launch_bounds__(256) void roi_pool_bilinear_wmma(
    const float* __restrict__ img, const int* __restrict__ rois,
    float* __restrict__ out) {
  // Stage-1 result: tmp[m=px (16 rows)][n = j*64 + c], n < 896. 57,344 B LDS.
  __shared__ float tmp[16 * 896];

  const int r    = blockIdx.x;
  const int lane = threadIdx.x & 31;
  const int wave = threadIdx.x >> 5;
  const int nw   = blockDim.x >> 5;

  const int rx = rois[r * 6 + 0];
  const int ry = rois[r * 6 + 1];
  const int rz = rois[r * 6 + 2];
  const int rw = rois[r * 6 + 3];
  const int rh = rois[r * 6 + 4];
  const int zc = clampi(rz, 0, ZD - 1);

  const int half = (lane < 16) ? 0 : 1;  // lane half selects K pairs / M rows
  const int mrow = lane & 15;

  // ---------------- Stage 1 ----------------
  // tmp[px][j*64+c] = sum_k WX[px,k] * img[Xc[k], Yc[j], zc, c]
  // 56 N-tiles of 16 (j = nt/4, c-base = (nt%4)*16), K = 16 in 4 chunks of 4.
  for (int nt = wave; nt < 56; nt += nw) {
    const int j     = nt >> 2;
    const int cbase = (nt & 3) << 4;
    const int yj    = axcoord(j, rh, ry, YD);
    const int c     = cbase + mrow;
    v8f acc = {};
#pragma unroll
    for (int kc = 0; kc < 4; ++kc) {
      // A frag (16x4 fp32): lanes 0-15 hold K = {4kc,4kc+1}; lanes 16-31 {4kc+2,4kc+3}
      v2f a, b;
      a.x = axweight(mrow, 4 * kc + 2 * half + 0, rw);
      a.y = axweight(mrow, 4 * kc + 2 * half + 1, rw);
      // B frag (4x16 fp32): VGPR0 rows K0|K1 across lane halves, VGPR1 rows K2|K3
      const int k0 = 4 * kc + half;
      const int k1 = k0 + 2;
      const float* p0 =
          img + (((size_t)axcoord(k0, rw, rx, XD) * YD + yj) * ZD + zc) * CD + c;
      const float* p1 =
          img + (((size_t)axcoord(k1, rw, rx, XD) * YD + yj) * ZD + zc) * CD + c;
      b.x = *p0;  // coalesced: 16 lanes cover 64 contiguous bytes of a channel row
      b.y = *p1;
      acc = __builtin_amdgcn_wmma_f32_16x16x4_f32(false, a, false, b, (short)0,
                                                  acc, false, false);
    }
    // D layout: VGPR q -> rows M=q (lanes 0-15) and M=q+8 (lanes 16-31)
    const int nidx  = j * 64 + cbase + mrow;
    const int mbase = half * 8;
#pragma unroll
    for (int q = 0; q < 8; ++q)
      tmp[(mbase + q) * 896 + nidx] = acc[q];
  }
  __syncthreads();

  // ---------------- Stage 2 ----------------
  // out[px, py, c] = sum_j WY[py,j] * tmp[px][j*64+c]
  // 28 N-tiles of 16 (px = nt/4, c-base = (nt%4)*16).
  for (int nt = wave; nt < 28; nt += nw) {
    const int px    = nt >> 2;
    const int cbase = (nt & 3) << 4;
    v8f acc = {};
#pragma unroll
    for (int kc = 0; kc < 4; ++kc) {
      v2f a, b;
      a.x = axweight(mrow, 4 * kc + 2 * half + 0, rh);
      a.y = axweight(mrow, 4 * kc + 2 * half + 1, rh);
      const int k0 = 4 * kc + half;
      const int k1 = k0 + 2;
      const int j0 = k0 > 13 ? 13 : k0;  // padded K rows: weight is 0 anyway
      const int j1 = k1 > 13 ? 13 : k1;
      b.x = tmp[px * 896 + j0 * 64 + cbase + mrow];
      b.y = tmp[px * 896 + j1 * 64 + cbase + mrow];
      acc = __builtin_amdgcn_wmma_f32_16x16x4_f32(false, a, false, b, (short)0,
                                                  acc, false, false);
    }
    const int mbase = half * 8;
#pragma unroll
    for (int q = 0; q < 8; ++q) {
      const int py = mbase + q;
      if (py < PP)
        out[(((size_t)r * PP + px) * PP + py) * CD + cbase + mrow] = acc[q];
    }
  }
}

extern "C" void kernel_launch(void* const* d_in, const int* in_sizes, int n_in,
                              void* d_out, int out_size, void* d_ws, size_t ws_size,
                              hipStream_t stream) {
  const float* img  = (const float*)d_in[0];
  const int*   rois = (const int*)d_in[1];
  float*       out  = (float*)d_out;
  (void)n_in; (void)out_size; (void)d_ws; (void)ws_size;

  const int R = in_sizes[1] / 6;  // (1, R, 6) int32 ROIs
  roi_pool_bilinear_wmma<<<dim3(R), dim3(256), 0, stream>>>(img, rois, out);
}